// BM3D_43327630082349
// MI455X (gfx1250) — compile-verified
//
#include <hip/hip_runtime.h>
#include <hip/hip_bf16.h>
#include <math.h>

// ---------------- problem constants (must match reference) ----------------
#define PSZ    8            // patch size
#define STR    4            // reference stride
#define KG     16           // group size
#define HP     377          // Hp = Wp = 384 - 8 + 1
#define NR     95           // # reference positions per dim (0..376 step 4)
#define NG     (NR * NR)    // 9025 groups
#define NC     81           // 9x9 search candidates
#define IMW    384
#define IMN    (IMW * IMW)  // 147456
#define LAMF   2.7f

typedef float v2f __attribute__((ext_vector_type(2)));
typedef float v8f __attribute__((ext_vector_type(8)));

// 8-point orthonormal DCT-II matrix (rows = frequency)
__device__ __constant__ float cD8[8][8] = {
  { 0.35355339f,  0.35355339f,  0.35355339f,  0.35355339f,  0.35355339f,  0.35355339f,  0.35355339f,  0.35355339f},
  { 0.49039264f,  0.41573481f,  0.27778512f,  0.09754516f, -0.09754516f, -0.27778512f, -0.41573481f, -0.49039264f},
  { 0.46193977f,  0.19134172f, -0.19134172f, -0.46193977f, -0.46193977f, -0.19134172f,  0.19134172f,  0.46193977f},
  { 0.41573481f, -0.09754516f, -0.49039264f, -0.27778512f,  0.27778512f,  0.49039264f,  0.09754516f, -0.41573481f},
  { 0.35355339f, -0.35355339f, -0.35355339f,  0.35355339f,  0.35355339f, -0.35355339f, -0.35355339f,  0.35355339f},
  { 0.27778512f, -0.49039264f,  0.09754516f,  0.41573481f, -0.41573481f, -0.09754516f,  0.49039264f, -0.27778512f},
  { 0.19134172f, -0.46193977f,  0.46193977f, -0.19134172f, -0.19134172f,  0.46193977f, -0.46193977f,  0.19134172f},
  { 0.09754516f, -0.27778512f,  0.41573481f, -0.49039264f,  0.49039264f, -0.41573481f,  0.27778512f, -0.09754516f}
};

// 16x16 orthonormal Hadamard (Sylvester): H(i,j) = (-1)^popc(i&j) / 4
static __device__ __forceinline__ float h16v(int i, int j) {
  return (__popc(i & j) & 1) ? -0.25f : 0.25f;
}

// ---------------- CDNA5 WMMA f32 16x16x4 ----------------
static __device__ __forceinline__ v8f wmma4(v2f a, v2f b, v8f c) {
#if __has_builtin(__builtin_amdgcn_wmma_f32_16x16x4_f32)
  return __builtin_amdgcn_wmma_f32_16x16x4_f32(false, a, false, b, (short)0, c,
                                               false, false);
#else
  c[0] += a[0] * b[0];   // placeholder fallback (keeps compile-safe)
  return c;
#endif
}

// Precompute the per-lane H16 A-fragments once (invariant over tiles/passes).
//   A 16x4 f32 : m = lane&15, VGPR0 = K=k0, VGPR1 = K=k0+1, k0 = 4kk + 2*(lane>>4)
static __device__ __forceinline__ void had_afrag(int half, int col, v2f af[4]) {
#pragma unroll
  for (int kk = 0; kk < 4; ++kk) {
    const int k0 = kk * 4 + half * 2;
    af[kk].x = h16v(col, k0);
    af[kk].y = h16v(col, k0 + 1);
  }
}

// One 16x16 output tile (columns t4*16..t4*16+15) of  D = H16 * M  (M is 16x64
// in LDS, row = group index k).  Fragment layouts per CDNA5 ISA §7.12.2:
//   B  4x16 f32: n = lane&15, K striped like A
//   D 16x16 f32: row m = r + 8*(lane>>4), col n = lane&15
static __device__ __forceinline__ v8f had_tile(const float (*M)[64],
                                               const v2f af[4], int t4,
                                               int half, int col) {
  v8f acc = {0.f, 0.f, 0.f, 0.f, 0.f, 0.f, 0.f, 0.f};
#pragma unroll
  for (int kk = 0; kk < 4; ++kk) {
    const int k0 = kk * 4 + half * 2;
    v2f bv;
    bv.x = M[k0][t4 * 16 + col];
    bv.y = M[k0 + 1][t4 * 16 + col];
    acc = wmma4(af[kk], bv, acc);
  }
  return acc;
}

static __device__ __forceinline__ float wave_sum32(float v) {
#pragma unroll
  for (int off = 16; off > 0; off >>= 1) v += __shfl_xor(v, off, 32);
  return v;
}

// forward 2-D DCT of the 8x8 patch at (pi,pj):  C = D8 * G * D8^T  -> dst[64]
static __device__ __forceinline__ void dct_fwd_store(const float* __restrict__ img,
                                                     int pi, int pj,
                                                     float* __restrict__ dst) {
  float g[8][8];
#pragma unroll
  for (int i = 0; i < 8; ++i)
#pragma unroll
    for (int j = 0; j < 8; ++j)
      g[i][j] = img[(pi + i) * IMW + (pj + j)];
#pragma unroll
  for (int a = 0; a < 8; ++a) {
    float t[8];
#pragma unroll
    for (int c = 0; c < 8; ++c) {
      float s = 0.f;
#pragma unroll
      for (int b = 0; b < 8; ++b) s += cD8[a][b] * g[b][c];
      t[c] = s;
    }
#pragma unroll
    for (int d = 0; d < 8; ++d) {
      float s = 0.f;
#pragma unroll
      for (int c = 0; c < 8; ++c) s += t[c] * cD8[d][c];
      dst[a * 8 + d] = s;
    }
  }
}

// inverse 2-D DCT (G = D8^T * C * D8) + weighted scatter into num/den
static __device__ __forceinline__ void dct_inv_aggregate(const float* __restrict__ src,
                                                         int pi, int pj, float w,
                                                         float* __restrict__ num,
                                                         float* __restrict__ den) {
  float c[8][8];
#pragma unroll
  for (int a = 0; a < 8; ++a)
#pragma unroll
    for (int d = 0; d < 8; ++d) c[a][d] = src[a * 8 + d];
#pragma unroll
  for (int b = 0; b < 8; ++b) {
    float t[8];
#pragma unroll
    for (int d = 0; d < 8; ++d) {
      float s = 0.f;
#pragma unroll
      for (int a = 0; a < 8; ++a) s += cD8[a][b] * c[a][d];
      t[d] = s;
    }
#pragma unroll
    for (int j = 0; j < 8; ++j) {
      float s = 0.f;
#pragma unroll
      for (int d = 0; d < 8; ++d) s += t[d] * cD8[d][j];
      const int pix = (pi + b) * IMW + (pj + j);
      atomicAdd(&num[pix], w * s);
      atomicAdd(&den[pix], w);
    }
  }
}

// ---------------- small utility kernels ----------------
__global__ void k_i2f(const int* __restrict__ im, float* __restrict__ out, int n) {
  int i = blockIdx.x * blockDim.x + threadIdx.x;
  if (i < n) out[i] = (float)im[i];
}

__global__ void k_zero(float* __restrict__ p, int n) {
  int i = blockIdx.x * blockDim.x + threadIdx.x;
  if (i < n) p[i] = 0.f;
}

__global__ void k_norm(const float* __restrict__ num, const float* __restrict__ den,
                       float* __restrict__ out, int n) {
  int i = blockIdx.x * blockDim.x + threadIdx.x;
  if (i < n) out[i] = num[i] / fmaxf(den[i], 1e-8f);
}

// ---------------- block matching ----------------
// one block per reference patch; LDS-staged reference; 81 candidate distances
__global__ __launch_bounds__(128) void k_dist(const float* __restrict__ img,
                                              float* __restrict__ dist) {
  const int n = blockIdx.x;
  const int a = n / NR, b = n % NR;
  const int ri = a * STR, rj = b * STR;
  __shared__ float ref[64];
  const int t = threadIdx.x;
  if (t < 64) ref[t] = img[(ri + t / 8) * IMW + rj + t % 8];
  __syncthreads();
  if (t < NC) {
    const int oi = t / 9, oj = t % 9;
    int ci = ri - 12 + 3 * oi; ci = min(max(ci, 0), HP - 1);
    int cj = rj - 12 + 3 * oj; cj = min(max(cj, 0), HP - 1);
    float scc = 0.f, scr = 0.f, srr = 0.f;
#pragma unroll 8
    for (int e = 0; e < 64; ++e) {
      const float cv = img[(ci + e / 8) * IMW + cj + e % 8];
      const float rv = ref[e];
      scc += cv * cv; scr += cv * rv; srr += rv * rv;
    }
    dist[n * NC + t] = scc - 2.f * scr + srr;
  }
}

// top-16 smallest distances, ties -> lower candidate index (matches lax.top_k)
__global__ __launch_bounds__(64) void k_select(const float* __restrict__ dist,
                                               int* __restrict__ gidx) {
  const int n = blockIdx.x * 64 + threadIdx.x;
  if (n >= NG) return;
  float d[NC];
  for (int c = 0; c < NC; ++c) d[c] = dist[n * NC + c];
  const int a = n / NR, b = n % NR;
  const int ri = a * STR, rj = b * STR;
  for (int k = 0; k < KG; ++k) {
    int bi = 0; float bv = d[0];
    for (int c = 1; c < NC; ++c)
      if (d[c] < bv) { bv = d[c]; bi = c; }
    d[bi] = __builtin_inff();
    int ci = ri - 12 + 3 * (bi / 9); ci = min(max(ci, 0), HP - 1);
    int cj = rj - 12 + 3 * (bi % 9); cj = min(max(cj, 0), HP - 1);
    gidx[n * KG + k] = ci * HP + cj;
  }
}

// ---------------- step 1: hard-threshold collaborative filtering ----------------
__global__ __launch_bounds__(32) void k_group_ht(const float* __restrict__ img,
                                                 const int* __restrict__ gidx,
                                                 const int* __restrict__ varp,
                                                 float* __restrict__ num,
                                                 float* __restrict__ den) {
  const int n = blockIdx.x;
  const int lane = threadIdx.x;
  const float sigma2 = (float)(*varp);
  const float thr = LAMF * sqrtf(sigma2);

  __shared__ float Cm[16][64];  // k-major DCT coeffs / inverse-Hadamard output
  __shared__ float Tm[16][64];  // g-major thresholded transform coeffs

  const int pk = gidx[n * KG + (lane & 15)];
  const int pi = pk / HP, pj = pk % HP;
  if (lane < 16) dct_fwd_store(img, pi, pj, Cm[lane]);
  __syncthreads();

  const int half = lane >> 4, col = lane & 15;
  v2f af[4];
  had_afrag(half, col, af);   // H16 A-fragments, reused by every WMMA below

  float nnz = 0.f;
#pragma unroll
  for (int t4 = 0; t4 < 4; ++t4) {
    v8f acc = had_tile(Cm, af, t4, half, col);   // forward Hadamard (WMMA)
#pragma unroll
    for (int r = 0; r < 8; ++r) {
      float v = acc[r];
      bool keep = fabsf(v) > thr;
      if (t4 == 0 && r == 0 && lane == 0) keep = true;  // DC (g=0,a=0,d=0)
      nnz += keep ? 1.f : 0.f;
      Tm[half * 8 + r][t4 * 16 + col] = keep ? v : 0.f;
    }
  }
  const float w = 1.f / (sigma2 * fmaxf(wave_sum32(nnz), 1.f));
  __syncthreads();
#pragma unroll
  for (int t4 = 0; t4 < 4; ++t4) {
    v8f acc = had_tile(Tm, af, t4, half, col);   // inverse Hadamard (H symmetric)
#pragma unroll
    for (int r = 0; r < 8; ++r) Cm[half * 8 + r][t4 * 16 + col] = acc[r];
  }
  __syncthreads();
  if (lane < 16) dct_inv_aggregate(Cm[lane], pi, pj, w, num, den);
}

// ---------------- step 2: Wiener collaborative filtering ----------------
__global__ __launch_bounds__(32) void k_group_wie(const float* __restrict__ noisy,
                                                  const float* __restrict__ basic,
                                                  const int* __restrict__ gidx,
                                                  const int* __restrict__ varp,
                                                  float* __restrict__ num,
                                                  float* __restrict__ den) {
  const int n = blockIdx.x;
  const int lane = threadIdx.x;
  const float sigma2 = (float)(*varp);

  __shared__ float Ab[16][64];  // basic-patch DCT coeffs (k-major)
  __shared__ float An[16][64];  // noisy-patch DCT coeffs (k-major)
  __shared__ float Tm[16][64];  // Wiener-filtered transform coeffs (g-major)

  const int half = lane >> 4, col = lane & 15;
  const int pk = gidx[n * KG + col];
  const int pi = pk / HP, pj = pk % HP;
  if (lane < 16) dct_fwd_store(basic, pi, pj, Ab[col]);
  else           dct_fwd_store(noisy, pi, pj, An[col]);
  __syncthreads();

  v2f af[4];
  had_afrag(half, col, af);   // shared across all three Hadamard passes

  float sumw2 = 0.f;
#pragma unroll
  for (int t4 = 0; t4 < 4; ++t4) {
    v8f cb = had_tile(Ab, af, t4, half, col);    // basic coeffs (WMMA)
    v8f cn = had_tile(An, af, t4, half, col);    // noisy coeffs (WMMA)
#pragma unroll
    for (int r = 0; r < 8; ++r) {
      const float b2 = cb[r] * cb[r];
      const float wn = b2 / (b2 + sigma2);
      sumw2 += wn * wn;
      Tm[half * 8 + r][t4 * 16 + col] = wn * cn[r];
    }
  }
  const float w = 1.f / (sigma2 * fmaxf(wave_sum32(sumw2), 1e-8f));
  __syncthreads();
#pragma unroll
  for (int t4 = 0; t4 < 4; ++t4) {
    v8f acc = had_tile(Tm, af, t4, half, col);   // inverse Hadamard
#pragma unroll
    for (int r = 0; r < 8; ++r) Ab[half * 8 + r][t4 * 16 + col] = acc[r];
  }
  __syncthreads();
  if (lane < 16) dct_inv_aggregate(Ab[lane], pi, pj, w, num, den);
}

// ---------------- host-side launch ----------------
extern "C" void kernel_launch(void* const* d_in, const int* in_sizes, int n_in,
                              void* d_out, int out_size, void* d_ws, size_t ws_size,
                              hipStream_t stream) {
  (void)in_sizes; (void)n_in; (void)out_size; (void)ws_size;
  const int* im   = (const int*)d_in[0];  // (1,1,384,384) int32
  const int* varp = (const int*)d_in[1];  // scalar variance (int32)
  float* out = (float*)d_out;             // (1,1,384,384) float32

  float* wsf   = (float*)d_ws;
  float* imgf  = wsf;                          // IMN
  float* basic = imgf + IMN;                   // IMN
  float* num   = basic + IMN;                  // IMN
  float* den   = num + IMN;                    // IMN   (contiguous with num)
  float* dist  = den + IMN;                    // NG*NC
  int*   gidx  = (int*)(dist + (size_t)NG * NC);  // NG*KG

  k_i2f<<<(IMN + 255) / 256, 256, 0, stream>>>(im, imgf, IMN);

  // ---- step 1: hard threshold ----
  k_dist<<<NG, 128, 0, stream>>>(imgf, dist);
  k_select<<<(NG + 63) / 64, 64, 0, stream>>>(dist, gidx);
  k_zero<<<(2 * IMN + 255) / 256, 256, 0, stream>>>(num, 2 * IMN);
  k_group_ht<<<NG, 32, 0, stream>>>(imgf, gidx, varp, num, den);
  k_norm<<<(IMN + 255) / 256, 256, 0, stream>>>(num, den, basic, IMN);

  // ---- step 2: Wiener ----
  k_dist<<<NG, 128, 0, stream>>>(basic, dist);
  k_select<<<(NG + 63) / 64, 64, 0, stream>>>(dist, gidx);
  k_zero<<<(2 * IMN + 255) / 256, 256, 0, stream>>>(num, 2 * IMN);
  k_group_wie<<<NG, 32, 0, stream>>>(imgf, basic, gidx, varp, num, den);
  k_norm<<<(IMN + 255) / 256, 256, 0, stream>>>(num, den, out, IMN);
}